// ExplainerGC_84722524881038
// MI455X (gfx1250) — compile-verified
//
#include <hip/hip_runtime.h>
#include <hip/hip_bf16.h>
#include <math.h>

#define N_NODES 8192
#define N_EDGES 81920
#define N_EDGE2 1310720
#define E_TOT   (N_EDGE2 + N_EDGES)   // expanded edges incl. self loops
#define D_GAT   64
#define HSZ     262144u
#define HMASK   (HSZ - 1u)

typedef __attribute__((ext_vector_type(2))) float v2f;
typedef __attribute__((ext_vector_type(8))) float v8f;

// ---------- helpers ----------
__device__ __forceinline__ float wave_sum(float v) {
    for (int o = 16; o > 0; o >>= 1) v += __shfl_down(v, o, 32);
    return v;
}
__device__ __forceinline__ float sigmoidf_(float x) { return 1.0f / (1.0f + expf(-x)); }

// order-preserving float<->uint map for atomicMax on floats
__device__ __forceinline__ unsigned f2ord(float f) {
    unsigned u = __float_as_uint(f);
    return (u & 0x80000000u) ? ~u : (u | 0x80000000u);
}
__device__ __forceinline__ float ord2f(unsigned u) {
    return (u & 0x80000000u) ? __uint_as_float(u & 0x7fffffffu) : __uint_as_float(~u);
}

__device__ __forceinline__ unsigned hash_u32(unsigned k) {
    k *= 0x9E3779B1u; k ^= k >> 16; k *= 0x85EBCA77u; k ^= k >> 13;
    return k;
}

// ---------- fills ----------
__global__ void k_fill_f32(float* p, float v, int n) {
    int i = blockIdx.x * blockDim.x + threadIdx.x;
    if (i < n) p[i] = v;
}
__global__ void k_fill_u32(unsigned* p, unsigned v, int n) {
    int i = blockIdx.x * blockDim.x + threadIdx.x;
    if (i < n) p[i] = v;
}
__global__ void k_bias_init(float* gat_out, const float* __restrict__ bias, int n) {
    int i = blockIdx.x * blockDim.x + threadIdx.x;
    if (i < n) gat_out[i] = bias[i & 63];
}

// ---------- GEMM1: hp = concat(embed[row], embed[col]) @ gat_W  (81920x256x64, f32 WMMA) ----------
// block = 128 threads = 4 waves; each wave computes a 16x64 tile; A gathered directly,
// B (gat_W, 256x64) staged in LDS with stride 68 to break bank conflicts.
__global__ void k_hp(const float* __restrict__ embed, const float* __restrict__ gatW,
                     const int* __restrict__ eidx, float* __restrict__ hp) {
    extern __shared__ float sW[];                       // [256][68]
    const int tid = threadIdx.x;
    for (int i = tid; i < 256 * 64; i += 128) {
        int r = i >> 6, c = i & 63;
        sW[r * 68 + c] = gatW[i];
    }
    __syncthreads();

    const int wave = tid >> 5, lane = tid & 31;
    const int lo = lane & 15, g = lane >> 4;
    const int m0 = blockIdx.x * 64 + wave * 16;
    const int e  = m0 + lo;
    const int rIdx = eidx[e];
    const int cIdx = eidx[N_EDGES + e];
    const float* rowPtr = embed + (size_t)rIdx * 128;
    const float* colPtr = embed + (size_t)cIdx * 128;

    v8f acc0 = {}, acc1 = {}, acc2 = {}, acc3 = {};
    for (int k0 = 0; k0 < 256; k0 += 4) {
        const int kk = k0 + 2 * g;                       // lane's K pair base
        const float* ap = (kk < 128) ? (rowPtr + kk) : (colPtr + (kk - 128));
        v2f a; a.x = ap[0]; a.y = ap[1];
        const float* b0 = sW + kk * 68;
        const float* b1 = sW + (kk + 1) * 68;
        v2f b;
        b.x = b0[lo];      b.y = b1[lo];
        acc0 = __builtin_amdgcn_wmma_f32_16x16x4_f32(false, a, false, b, (short)0, acc0, false, false);
        b.x = b0[16 + lo]; b.y = b1[16 + lo];
        acc1 = __builtin_amdgcn_wmma_f32_16x16x4_f32(false, a, false, b, (short)0, acc1, false, false);
        b.x = b0[32 + lo]; b.y = b1[32 + lo];
        acc2 = __builtin_amdgcn_wmma_f32_16x16x4_f32(false, a, false, b, (short)0, acc2, false, false);
        b.x = b0[48 + lo]; b.y = b1[48 + lo];
        acc3 = __builtin_amdgcn_wmma_f32_16x16x4_f32(false, a, false, b, (short)0, acc3, false, false);
    }
    float* outp = hp + (size_t)m0 * 64;
#pragma unroll
    for (int j = 0; j < 8; ++j) {
        int M = j + 8 * g;
        outp[(size_t)M * 64 +      lo] = acc0[j];
        outp[(size_t)M * 64 + 16 + lo] = acc1[j];
        outp[(size_t)M * 64 + 32 + lo] = acc2[j];
        outp[(size_t)M * 64 + 48 + lo] = acc3[j];
    }
}

// ---------- a_s / a_d : wave-per-edge dot products ----------
__global__ void k_asd(const float* __restrict__ hp, const float* __restrict__ att_src,
                      const float* __restrict__ att_dst, float* __restrict__ a_s,
                      float* __restrict__ a_d) {
    int t = blockIdx.x * blockDim.x + threadIdx.x;
    int e = t >> 5, lane = t & 31;
    const float* h = hp + (size_t)e * 64;
    float h0 = h[lane], h1 = h[lane + 32];
    float s = h0 * att_src[lane] + h1 * att_src[lane + 32];
    float d = h0 * att_dst[lane] + h1 * att_dst[lane + 32];
    s = wave_sum(s); d = wave_sum(d);
    if (lane == 0) { a_s[e] = s; a_d[e] = d; }
}

// ---------- attention: alpha + segment max ----------
__global__ void k_alpha(const float* __restrict__ a_s, const float* __restrict__ a_d,
                        const int* __restrict__ nei, float* __restrict__ alpha,
                        unsigned* __restrict__ amax_u) {
    int i = blockIdx.x * blockDim.x + threadIdx.x;
    if (i >= E_TOT) return;
    int src, dst;
    if (i < N_EDGE2) { src = nei[i]; dst = nei[N_EDGE2 + i]; }
    else             { src = dst = i - N_EDGE2; }
    float xv = a_s[src] + a_d[dst];
    float al = xv > 0.0f ? xv : 0.2f * xv;               // leaky_relu 0.2
    alpha[i] = al;
    atomicMax(&amax_u[dst], f2ord(al));
}

// ---------- exp & segment sum (in-place alpha -> ealpha) ----------
__global__ void k_exps(const int* __restrict__ nei, const unsigned* __restrict__ amax_u,
                       float* __restrict__ alpha, float* __restrict__ denom) {
    int i = blockIdx.x * blockDim.x + threadIdx.x;
    if (i >= E_TOT) return;
    int dst = (i < N_EDGE2) ? nei[N_EDGE2 + i] : (i - N_EDGE2);
    float ev = expf(alpha[i] - ord2f(amax_u[dst]));
    alpha[i] = ev;
    atomicAdd(&denom[dst], ev);
}

// ---------- gat_out scatter: 16 threads per expanded edge, float4 chunks ----------
__global__ void k_scatter(const float* __restrict__ ealpha, const float* __restrict__ denom,
                          const int* __restrict__ nei, const float* __restrict__ hp,
                          float* __restrict__ gat_out) {
    int t = blockIdx.x * blockDim.x + threadIdx.x;       // E_TOT*16 threads
    int i = t >> 4, c4 = (t & 15) * 4;
    int src, dst;
    if (i < N_EDGE2) { src = nei[i]; dst = nei[N_EDGE2 + i]; }
    else             { src = dst = i - N_EDGE2; }
    float attn = ealpha[i] / denom[dst];
    const float4 v = *(const float4*)(hp + (size_t)src * 64 + c4);
    float* o = gat_out + (size_t)dst * 64 + c4;
    atomicAdd(o + 0, attn * v.x);
    atomicAdd(o + 1, attn * v.y);
    atomicAdd(o + 2, attn * v.z);
    atomicAdd(o + 3, attn * v.w);
}

// ---------- gate = concrete_sample(sigmoid((log u - log1p(-u) + gat_out@lin_W + b)/tmp)) ----------
__global__ void k_gate(const float* __restrict__ gat_out, const float* __restrict__ lin_W,
                       const float* __restrict__ lin_b, const float* __restrict__ noise,
                       const float* __restrict__ tmp, float* __restrict__ gate) {
    int t = blockIdx.x * blockDim.x + threadIdx.x;
    int e = t >> 5, lane = t & 31;
    const float* go = gat_out + (size_t)e * 64;
    float s = go[lane] * lin_W[lane] + go[lane + 32] * lin_W[lane + 32];
    s = wave_sum(s);
    if (lane == 0) {
        float la = s + lin_b[0];
        float u  = noise[e];
        gate[e]  = sigmoidf_((logf(u) - log1pf(-u) + la) / tmp[0]);
    }
}

// ---------- sparse replacement of the dense NxN adjacency: hash (r*8192+c) -> (count, sum gate) ----------
__global__ void k_hash_insert(const int* __restrict__ eidx, const float* __restrict__ gate,
                              unsigned* __restrict__ hk, unsigned* __restrict__ hc,
                              float* __restrict__ hg) {
    int e = blockIdx.x * blockDim.x + threadIdx.x;
    if (e >= N_EDGES) return;
    unsigned key = (unsigned)eidx[e] * 8192u + (unsigned)eidx[N_EDGES + e] + 1u;
    unsigned h = hash_u32(key) & HMASK;
    while (true) {
        unsigned prev = atomicCAS(&hk[h], 0u, key);
        if (prev == 0u || prev == key) break;
        h = (h + 1u) & HMASK;
    }
    atomicAdd(&hc[h], 1u);
    atomicAdd(&hg[h], gate[e]);
}

__device__ __forceinline__ int hash_lookup(const unsigned* hk, unsigned key) {
    unsigned h = hash_u32(key) & HMASK;
    while (true) {
        unsigned k = hk[h];
        if (k == key) return (int)h;
        if (k == 0u)  return -1;
        h = (h + 1u) & HMASK;
    }
}

// edge_mask[e] = cnt(r,c) * 0.5*(gsum(r,c)+gsum(c,r));  w = sigmoid(edge_mask);  deg[col]+=w
__global__ void k_edge_w(const int* __restrict__ eidx, const unsigned* __restrict__ hk,
                         const unsigned* __restrict__ hc, const float* __restrict__ hg,
                         float* __restrict__ wbuf, float* __restrict__ deg) {
    int e = blockIdx.x * blockDim.x + threadIdx.x;
    if (e >= N_EDGES) return;
    unsigned r = (unsigned)eidx[e], c = (unsigned)eidx[N_EDGES + e];
    int h1 = hash_lookup(hk, r * 8192u + c + 1u);        // exists (this edge inserted it)
    int h2 = hash_lookup(hk, c * 8192u + r + 1u);
    float gs2 = (h2 >= 0) ? hg[h2] : 0.0f;
    float em  = (float)hc[h1] * 0.5f * (hg[h1] + gs2);
    float w   = sigmoidf_(em);
    wbuf[e] = w;
    atomicAdd(&deg[c], w);
}

// ---------- GCN message scatter: wave per edge, msg[col] += w*xh[row] (128 f32) ----------
__global__ void k_msg(const float* __restrict__ xh, const float* __restrict__ wbuf,
                      const int* __restrict__ eidx, float* __restrict__ msg) {
    int t = blockIdx.x * blockDim.x + threadIdx.x;
    int e = t >> 5, lane = t & 31;
    int r = eidx[e], c = eidx[N_EDGES + e];
    float we = wbuf[e];
    const float4 xv = *(const float4*)(xh + (size_t)r * 128 + lane * 4);
    float* o = msg + (size_t)c * 128 + lane * 4;
    atomicAdd(o + 0, we * xv.x);
    atomicAdd(o + 1, we * xv.y);
    atomicAdd(o + 2, we * xv.z);
    atomicAdd(o + 3, we * xv.w);
}

// ---------- GCN GEMM: y = relu(((xh+msg)/deg) @ W)  (8192x128x128, f32 WMMA) ----------
__global__ void k_gcn(const float* __restrict__ xh, const float* __restrict__ msg,
                      const float* __restrict__ deg, const float* __restrict__ W,
                      float* __restrict__ y) {
    extern __shared__ float smem[];
    float* sW = smem;                                    // [128][132]
    float* sA = smem + 128 * 132;                        // [4][16][132]
    const int tid = threadIdx.x;
    for (int i = tid; i < 128 * 128; i += 128) {
        int r = i >> 7, c = i & 127;
        sW[r * 132 + c] = W[i];
    }
    const int wave = tid >> 5, lane = tid & 31;
    const int lo = lane & 15, g = lane >> 4;
    const int m0 = blockIdx.x * 64 + wave * 16;
    float* myA = sA + wave * 16 * 132;
    for (int r = 0; r < 16; ++r) {
        int m = m0 + r;
        float dinv = 1.0f / deg[m];
        int c = lane * 4;
        const float4 xv = *(const float4*)(xh  + (size_t)m * 128 + c);
        const float4 mv = *(const float4*)(msg + (size_t)m * 128 + c);
        float* dst = myA + r * 132 + c;
        dst[0] = (xv.x + mv.x) * dinv;
        dst[1] = (xv.y + mv.y) * dinv;
        dst[2] = (xv.z + mv.z) * dinv;
        dst[3] = (xv.w + mv.w) * dinv;
    }
    __syncthreads();

    v8f acc[8];
#pragma unroll
    for (int t2 = 0; t2 < 8; ++t2) acc[t2] = (v8f){};
    for (int k0 = 0; k0 < 128; k0 += 4) {
        const int kk = k0 + 2 * g;
        v2f a; a.x = myA[lo * 132 + kk]; a.y = myA[lo * 132 + kk + 1];
#pragma unroll
        for (int t2 = 0; t2 < 8; ++t2) {
            v2f b;
            b.x = sW[kk * 132 + t2 * 16 + lo];
            b.y = sW[(kk + 1) * 132 + t2 * 16 + lo];
            acc[t2] = __builtin_amdgcn_wmma_f32_16x16x4_f32(false, a, false, b, (short)0, acc[t2], false, false);
        }
    }
    float* outp = y + (size_t)m0 * 128;
#pragma unroll
    for (int t2 = 0; t2 < 8; ++t2)
#pragma unroll
        for (int j = 0; j < 8; ++j) {
            int M = j + 8 * g;
            outp[(size_t)M * 128 + t2 * 16 + lo] = fmaxf(acc[t2][j], 0.0f);
        }
}

// ---------- mean pool partials ----------
__global__ void k_pool(const float* __restrict__ h3, float* __restrict__ gvec) {
    int c = threadIdx.x, b = blockIdx.x;
    const float* p = h3 + (size_t)b * 64 * 128 + c;
    float s = 0.0f;
    for (int r = 0; r < 64; ++r) s += p[r * 128];
    atomicAdd(&gvec[c], s);
}

// ---------- classifier + softmax ----------
__global__ void k_final(const float* __restrict__ gvec, const float* __restrict__ Wc,
                        float* __restrict__ outp) {
    int lane = threadIdx.x;
    float p0 = 0.0f, p1 = 0.0f;
    for (int c = lane; c < 128; c += 32) {
        float gc = gvec[c] * (1.0f / 8192.0f);
        p0 += gc * Wc[c * 2 + 0];
        p1 += gc * Wc[c * 2 + 1];
    }
    p0 = wave_sum(p0); p1 = wave_sum(p1);
    if (lane == 0) {
        float m = fmaxf(p0, p1);
        float e0 = expf(p0 - m), e1 = expf(p1 - m);
        float inv = 1.0f / (e0 + e1);
        outp[0] = e0 * inv;
        outp[1] = e1 * inv;
    }
}

// ---------- host orchestration ----------
extern "C" void kernel_launch(void* const* d_in, const int* in_sizes, int n_in,
                              void* d_out, int out_size, void* d_ws, size_t ws_size,
                              hipStream_t stream) {
    const float* x       = (const float*)d_in[0];
    const float* embed   = (const float*)d_in[1];
    const float* noise   = (const float*)d_in[2];
    const float* tmp     = (const float*)d_in[3];
    const float* gat_W   = (const float*)d_in[4];
    const float* att_src = (const float*)d_in[5];
    const float* att_dst = (const float*)d_in[6];
    const float* gat_b   = (const float*)d_in[7];
    const float* lin_W   = (const float*)d_in[8];
    const float* lin_b   = (const float*)d_in[9];
    const float* W1      = (const float*)d_in[10];
    const float* W2      = (const float*)d_in[11];
    const float* W3      = (const float*)d_in[12];
    const float* Wc      = (const float*)d_in[13];
    const int*   eidx    = (const int*)d_in[14];
    const int*   nei     = (const int*)d_in[15];
    float* outp = (float*)d_out;

    char* p = (char*)d_ws;
    auto alloc = [&](size_t bytes) { void* r = (void*)p; p += (bytes + 255) & ~(size_t)255; return r; };
    float*    hp      = (float*)   alloc((size_t)N_EDGES * 64 * 4);
    float*    a_s     = (float*)   alloc((size_t)N_EDGES * 4);
    float*    a_d     = (float*)   alloc((size_t)N_EDGES * 4);
    unsigned* amax_u  = (unsigned*)alloc((size_t)N_EDGES * 4);
    float*    denom   = (float*)   alloc((size_t)N_EDGES * 4);
    float*    alpha   = (float*)   alloc((size_t)E_TOT * 4);
    float*    gat_out = (float*)   alloc((size_t)N_EDGES * 64 * 4);
    float*    gate    = (float*)   alloc((size_t)N_EDGES * 4);
    unsigned* hk      = (unsigned*)alloc((size_t)HSZ * 4);
    unsigned* hc      = (unsigned*)alloc((size_t)HSZ * 4);
    float*    hg      = (float*)   alloc((size_t)HSZ * 4);
    float*    wbuf    = (float*)   alloc((size_t)N_EDGES * 4);
    float*    deg     = (float*)   alloc((size_t)N_NODES * 4);
    float*    msg     = (float*)   alloc((size_t)N_NODES * 128 * 4);
    float*    hA      = (float*)   alloc((size_t)N_NODES * 128 * 4);
    float*    hB      = (float*)   alloc((size_t)N_NODES * 128 * 4);
    float*    gvec    = (float*)   alloc(128 * 4);

#define GRID(n, b) dim3(((n) + (b) - 1) / (b))
    // init
    k_fill_u32<<<GRID(N_EDGES, 256), 256, 0, stream>>>(amax_u, 0u, N_EDGES);
    k_fill_f32<<<GRID(N_EDGES, 256), 256, 0, stream>>>(denom, 0.0f, N_EDGES);
    k_bias_init<<<GRID(N_EDGES * 64, 256), 256, 0, stream>>>(gat_out, gat_b, N_EDGES * 64);
    k_fill_f32<<<GRID(N_NODES, 256), 256, 0, stream>>>(deg, 1.0f, N_NODES);
    k_fill_u32<<<GRID((int)HSZ, 256), 256, 0, stream>>>(hk, 0u, (int)HSZ);
    k_fill_u32<<<GRID((int)HSZ, 256), 256, 0, stream>>>(hc, 0u, (int)HSZ);
    k_fill_f32<<<GRID((int)HSZ, 256), 256, 0, stream>>>(hg, 0.0f, (int)HSZ);
    k_fill_f32<<<GRID(128, 128), 128, 0, stream>>>(gvec, 0.0f, 128);

    // GAT front end
    k_hp<<<dim3(N_EDGES / 64), dim3(128), 256 * 68 * 4, stream>>>(embed, gat_W, eidx, hp);
    k_asd<<<GRID(N_EDGES * 32, 256), 256, 0, stream>>>(hp, att_src, att_dst, a_s, a_d);
    k_alpha<<<GRID(E_TOT, 256), 256, 0, stream>>>(a_s, a_d, nei, alpha, amax_u);
    k_exps<<<GRID(E_TOT, 256), 256, 0, stream>>>(nei, amax_u, alpha, denom);
    k_scatter<<<GRID(E_TOT * 16, 256), 256, 0, stream>>>(alpha, denom, nei, hp, gat_out);
    k_gate<<<GRID(N_EDGES * 32, 256), 256, 0, stream>>>(gat_out, lin_W, lin_b, noise, tmp, gate);

    // sparse masked adjacency -> per-edge weight + degree
    k_hash_insert<<<GRID(N_EDGES, 256), 256, 0, stream>>>(eidx, gate, hk, hc, hg);
    k_edge_w<<<GRID(N_EDGES, 256), 256, 0, stream>>>(eidx, hk, hc, hg, wbuf, deg);

    // 3-layer edge-weighted GCN
    const size_t gcnLds = (128 * 132 + 4 * 16 * 132) * 4;
    const float* layerIn[3] = { x, hA, hB };
    const float* layerW[3]  = { W1, W2, W3 };
    float*       layerOut[3]= { hA, hB, hA };
    for (int l = 0; l < 3; ++l) {
        k_fill_f32<<<GRID(N_NODES * 128, 256), 256, 0, stream>>>(msg, 0.0f, N_NODES * 128);
        k_msg<<<GRID(N_EDGES * 32, 256), 256, 0, stream>>>(layerIn[l], wbuf, eidx, msg);
        k_gcn<<<dim3(N_NODES / 64), dim3(128), gcnLds, stream>>>(layerIn[l], msg, deg, layerW[l], layerOut[l]);
    }

    // pool + classify
    k_pool<<<dim3(N_NODES / 64), dim3(128), 0, stream>>>(hA, gvec);
    k_final<<<dim3(1), dim3(32), 0, stream>>>(gvec, Wc, outp);
#undef GRID
}